// GCN_new_16389595202325
// MI455X (gfx1250) — compile-verified
//
#include <hip/hip_runtime.h>

typedef __attribute__((ext_vector_type(2))) float v2f;
typedef __attribute__((ext_vector_type(8))) float v8f;

#define GCN_D 512   // INP_DIM == HID == 512 in the reference

// ---------------------------------------------------------------------------
// C[m][n] = act( sum_k A[m][k] * W[n][k] + bias[n] )
// One block (8 wave32) = 16 output rows x 512 cols.
// One wave = 16x64 slab = 4 accumulators of 16x16, fp32 WMMA 16x16x4.
// ---------------------------------------------------------------------------
template <bool PRELU>
__global__ __launch_bounds__(256, 2)
void gemm_bias_act_wmma(const float* __restrict__ A,      // [nrows, 512]
                        const float* __restrict__ W,      // [512, 512], row n = output col n
                        const float* __restrict__ bias,   // [512]
                        const float* __restrict__ slope_p,
                        float* __restrict__ C,            // [nrows, 512]
                        int nrows)
{
    constexpr int D = GCN_D;
    const int lane = threadIdx.x & 31;
    const int wave = threadIdx.x >> 5;
    const int l15  = lane & 15;
    const int kh   = lane >> 4;          // 0 -> K pair {0,1}; 1 -> K pair {2,3}
    const int row0 = blockIdx.x * 16;
    const int n0   = wave * 64;

    // Clamp A row (keeps EXEC all-ones, as WMMA requires); stores are predicated.
    int arow = row0 + l15;
    if (arow >= nrows) arow = nrows - 1;

    const float* ap = A + (size_t)arow * D + 2 * kh;
    const float* wp[4];
#pragma unroll
    for (int j = 0; j < 4; ++j)
        wp[j] = W + (size_t)(n0 + j * 16 + l15) * D + 2 * kh;

    v8f acc[4];
#pragma unroll
    for (int j = 0; j < 4; ++j) acc[j] = (v8f){};

    for (int k = 0; k < D; k += 4) {
        const v2f a = *(const v2f*)(ap + k);
#pragma unroll
        for (int j = 0; j < 4; ++j) {
            const v2f b = *(const v2f*)(wp[j] + k);
            // (neg_a, A, neg_b, B, c_mod, C, reuse_a, reuse_b)
            acc[j] = __builtin_amdgcn_wmma_f32_16x16x4_f32(
                false, a, false, b, (short)0, acc[j], false, false);
        }
    }

    float sl = 0.0f;
    if (PRELU) sl = slope_p[0];

#pragma unroll
    for (int j = 0; j < 4; ++j) {
        const float bb = bias[n0 + j * 16 + l15];
#pragma unroll
        for (int r = 0; r < 8; ++r) {
            const int row = row0 + r + 8 * kh;   // C/D layout: lanes 16-31 hold M+8
            if (row < nrows) {
                float v = acc[j][r] + bb;
                if (PRELU) v = (v >= 0.0f) ? v : sl * v;
                C[(size_t)row * D + n0 + j * 16 + l15] = v;
            }
        }
    }
}

// ---------------------------------------------------------------------------
// out[r, :] += v * t[c, :]  via float4 gather + 4 f32 atomics. 128 lanes/edge.
// ---------------------------------------------------------------------------
__global__ __launch_bounds__(256)
void spmm_scatter(const float* __restrict__ t,
                  const float* __restrict__ vals,
                  const int* __restrict__ rows,
                  const int* __restrict__ cols,
                  float* __restrict__ out,
                  int nE)
{
    const long long tid = (long long)blockIdx.x * 256 + threadIdx.x;
    const long long e   = tid >> 7;
    if (e >= nE) return;
    const int ch = (int)(tid & 127) * 4;

    const float v = vals[e];
    const int r = rows[e];
    const int c = cols[e];

    const float4 tv = *(const float4*)(t + (size_t)c * GCN_D + ch);
    float* dst = out + (size_t)r * GCN_D + ch;
    atomicAdd(dst + 0, v * tv.x);
    atomicAdd(dst + 1, v * tv.y);
    atomicAdd(dst + 2, v * tv.z);
    atomicAdd(dst + 3, v * tv.w);
}

__global__ __launch_bounds__(256)
void zero4_kernel(float4* __restrict__ p, size_t n4)
{
    size_t i = (size_t)blockIdx.x * blockDim.x + threadIdx.x;
    const size_t stride = (size_t)gridDim.x * blockDim.x;
    const float4 z = make_float4(0.f, 0.f, 0.f, 0.f);
    for (; i < n4; i += stride) p[i] = z;
}

__global__ __launch_bounds__(256)
void prelu4_kernel(float4* __restrict__ p, const float* __restrict__ slope_p, size_t n4)
{
    const float sl = slope_p[0];
    size_t i = (size_t)blockIdx.x * blockDim.x + threadIdx.x;
    const size_t stride = (size_t)gridDim.x * blockDim.x;
    for (; i < n4; i += stride) {
        float4 v = p[i];
        v.x = (v.x >= 0.f) ? v.x : sl * v.x;
        v.y = (v.y >= 0.f) ? v.y : sl * v.y;
        v.z = (v.z >= 0.f) ? v.z : sl * v.z;
        v.w = (v.w >= 0.f) ? v.w : sl * v.w;
        p[i] = v;
    }
}

extern "C" void kernel_launch(void* const* d_in, const int* in_sizes, int n_in,
                              void* d_out, int out_size, void* d_ws, size_t ws_size,
                              hipStream_t stream)
{
    const float* AX = (const float*)d_in[0];
    const float* W0 = (const float*)d_in[1];
    const float* b0 = (const float*)d_in[2];
    const float* a0 = (const float*)d_in[3];
    const float* W1 = (const float*)d_in[4];
    const float* b1 = (const float*)d_in[5];
    const float* a1 = (const float*)d_in[6];
    const float* ev = (const float*)d_in[7];
    const int*   er = (const int*)d_in[8];
    const int*   ec = (const int*)d_in[9];
    float* out = (float*)d_out;

    const int D = GCN_D;
    const int N = in_sizes[0] / D;
    const int E = in_sizes[7];

    float* h = out;              // stage layer-0 activations in d_out (overwritten later)
    float* t = (float*)d_ws;     // layer-1 pre-aggregation lives in workspace (N*512*4 B)

    dim3 blk(256);
    const int gtiles = (N + 15) / 16;

    // Layer 0: h = PReLU(AX @ W0^T + b0, a0)
    gemm_bias_act_wmma<true><<<gtiles, blk, 0, stream>>>(AX, W0, b0, a0, h, N);
    // Layer 1 (pre-aggregation): t = h @ W1^T + b1
    gemm_bias_act_wmma<false><<<gtiles, blk, 0, stream>>>(h, W1, b1, a1, t, N);

    // Segment-sum destination must start at zero.
    const size_t n4 = (size_t)N * D / 4;
    zero4_kernel<<<2048, blk, 0, stream>>>((float4*)out, n4);

    // SpMM scatter: out[r] += v * t[c]
    const long long tot = (long long)E * 128;
    const int sblocks = (int)((tot + 255) / 256);
    spmm_scatter<<<sblocks, blk, 0, stream>>>(t, ev, er, ec, out, E);

    // Final PReLU(a1) in place.
    prelu4_kernel<<<2048, blk, 0, stream>>>((float4*)out, a1, n4);
}